// SoftDTW_25555055411351
// MI455X (gfx1250) — compile-verified
//
#include <hip/hip_runtime.h>
#include <hip/hip_bf16.h>
#include <cstdint>

typedef __attribute__((ext_vector_type(2))) float v2f;
typedef __attribute__((ext_vector_type(8))) float v8f;

#define B_ 32
#define N_ 512
#define M_ 512
#define K_ 64

// ---------------------------------------------------------------------------
// Kernel 0: row norms ||X[b,i]||^2 -> x2, ||Y[b,j]||^2 -> y2
// ---------------------------------------------------------------------------
__global__ __launch_bounds__(256) void sdtw_norms(const float* __restrict__ X,
                                                  const float* __restrict__ Y,
                                                  float* __restrict__ x2,
                                                  float* __restrict__ y2) {
  int t = blockIdx.x * blockDim.x + threadIdx.x;  // 0 .. 2*B*N-1
  const float* src;
  float* dst;
  int idx;
  if (t < B_ * N_) { src = X; dst = x2; idx = t; }
  else             { src = Y; dst = y2; idx = t - B_ * N_; }
  const float4* p = (const float4*)(src + (size_t)idx * K_);
  float s = 0.f;
#pragma unroll
  for (int k = 0; k < K_ / 4; ++k) {
    float4 v = p[k];
    s += v.x * v.x + v.y * v.y + v.z * v.z + v.w * v.w;
  }
  dst[idx] = s;
}

// ---------------------------------------------------------------------------
// Kernel 1: D[b,i,j] = x2[b,i] + y2[b,j] - 2 * (X_b Y_b^T)[i,j]
// One wave per 16x16 output tile, V_WMMA_F32_16X16X4_F32, K-loop of 16.
// A frag (16x4 f32): lane L<16 holds M=L, K={k0,k0+1}; lane L>=16 holds K={k0+2,k0+3}.
// B frag (4x16 f32): same pattern with N=lane&15.
// C/D (16x16 f32): vgpr r <-> M=r (lanes 0-15) / M=r+8 (lanes 16-31), N=lane&15.
// ---------------------------------------------------------------------------
__global__ __launch_bounds__(256) void sdtw_pairdist_wmma(
    const float* __restrict__ X, const float* __restrict__ Y,
    const float* __restrict__ x2, const float* __restrict__ y2,
    float* __restrict__ D) {
  int lane = threadIdx.x & 31;
  int wave = threadIdx.x >> 5;
  int tileLinear = blockIdx.x * 8 + wave;  // 0..1023 tiles within batch
  int b = blockIdx.y;
  int ti = tileLinear >> 5;   // tile row  0..31
  int tj = tileLinear & 31;   // tile col  0..31
  int l15 = lane & 15;
  int khalf = (lane >> 4) << 1;  // 0 or 2

  const float* arow = X + ((size_t)b * N_ + ti * 16 + l15) * K_ + khalf;
  const float* brow = Y + ((size_t)b * M_ + tj * 16 + l15) * K_ + khalf;

  v8f acc = {};
#pragma unroll
  for (int k0 = 0; k0 < K_; k0 += 4) {
    v2f av = *(const v2f*)(arow + k0);
    v2f bv = *(const v2f*)(brow + k0);
    // 8 args: (neg_a, A, neg_b, B, c_mod, C, reuse_a, reuse_b)
    acc = __builtin_amdgcn_wmma_f32_16x16x4_f32(false, av, false, bv,
                                                (short)0, acc, false, false);
  }

  int n = tj * 16 + l15;
  float y2v = y2[b * M_ + n];
  int mbase = ti * 16 + ((lane >> 4) ? 8 : 0);
  const float* x2p = x2 + b * N_ + mbase;
  float* Dp = D + ((size_t)b * N_ + mbase) * M_ + n;
#pragma unroll
  for (int r = 0; r < 8; ++r) {
    Dp[(size_t)r * M_] = x2p[r] + y2v - 2.0f * acc[r];
  }
}

// ---------------------------------------------------------------------------
// Kernel 2: soft-DTW anti-diagonal DP. One workgroup per batch, 512 threads
// (16 wave32s), three rolling diagonals in LDS, one barrier per diagonal.
// Thread t owns row i = t+1; at diagonal d it computes cell (i, j=d-i).
// softmin(a,b,c) = m - log(exp(m-a)+exp(m-b)+exp(m-c)), m = min; inf-safe.
// ---------------------------------------------------------------------------
__global__ __launch_bounds__(512) void sdtw_dp(const float* __restrict__ D,
                                               float* __restrict__ out) {
  __shared__ float buf[3][N_ + 1];
  const float INF = __builtin_inff();
  int b = blockIdx.x;
  int t = threadIdx.x;
  int i = t + 1;  // 1..N

  buf[0][i] = INF;   // diag d=0: only R[0,0]=0
  buf[1][i] = INF;   // diag d=1: all inf
  if (t == 0) { buf[0][0] = 0.f; buf[1][0] = INF; buf[2][0] = INF; }
  __syncthreads();

  const float* Db = D + (size_t)b * N_ * M_ + (size_t)(i - 1) * M_;
  int p2 = 0, p1 = 1, pc = 2;
  float val = INF;

  for (int d = 2; d <= N_ + M_; ++d) {
    int j = d - i;
    val = INF;
    if (j >= 1 && j <= M_) {
      float a  = buf[p2][i - 1];  // R[i-1, j-1]
      float bb = buf[p1][i - 1];  // R[i-1, j]
      float cc = buf[p1][i];      // R[i,   j-1]
      float mn = fminf(a, fminf(bb, cc));
      float sm = INF;
      if (mn < INF) {
        float s = __expf(mn - a) + __expf(mn - bb) + __expf(mn - cc);
        sm = mn - __logf(s);
      }
      val = Db[j - 1] + sm;
    }
    buf[pc][i] = val;
    if (t == 0) buf[pc][0] = INF;
    __syncthreads();
    int tmp = p2; p2 = p1; p1 = pc; pc = tmp;
  }
  if (i == N_) out[b] = val;  // R[N, M]
}

// ---------------------------------------------------------------------------
// Fallback (only if workspace can't hold D): fused kernel, X_b/Y_b staged in
// LDS (256KB < 320KB/WGP), distances recomputed per cell.
// ---------------------------------------------------------------------------
extern __shared__ float smem_f[];
__global__ __launch_bounds__(512) void sdtw_fused(const float* __restrict__ X,
                                                  const float* __restrict__ Y,
                                                  float* __restrict__ out) {
  float* Xs  = smem_f;                  // N_*K_
  float* Ys  = Xs + N_ * K_;            // M_*K_
  float* x2s = Ys + M_ * K_;            // N_
  float* y2s = x2s + N_;                // M_
  float (*buf)[N_ + 1] = (float (*)[N_ + 1])(y2s + M_);  // 3*(N_+1)
  const float INF = __builtin_inff();
  int b = blockIdx.x, t = threadIdx.x, i = t + 1;

  const float4* xr = (const float4*)(X + ((size_t)b * N_ + t) * K_);
  const float4* yr = (const float4*)(Y + ((size_t)b * M_ + t) * K_);
  float4* xsw = (float4*)(Xs + (size_t)t * K_);
  float4* ysw = (float4*)(Ys + (size_t)t * K_);
  float sx = 0.f, sy = 0.f;
#pragma unroll
  for (int k = 0; k < K_ / 4; ++k) {
    float4 a = xr[k]; float4 c = yr[k];
    xsw[k] = a; ysw[k] = c;
    sx += a.x * a.x + a.y * a.y + a.z * a.z + a.w * a.w;
    sy += c.x * c.x + c.y * c.y + c.z * c.z + c.w * c.w;
  }
  x2s[t] = sx; y2s[t] = sy;
  buf[0][i] = INF; buf[1][i] = INF;
  if (t == 0) { buf[0][0] = 0.f; buf[1][0] = INF; buf[2][0] = INF; }
  __syncthreads();

  const float4* xi = (const float4*)(Xs + (size_t)t * K_);
  int p2 = 0, p1 = 1, pc = 2;
  float val = INF;
  for (int d = 2; d <= N_ + M_; ++d) {
    int j = d - i;
    val = INF;
    if (j >= 1 && j <= M_) {
      float a  = buf[p2][i - 1];
      float bb = buf[p1][i - 1];
      float cc = buf[p1][i];
      float mn = fminf(a, fminf(bb, cc));
      float sm = INF;
      if (mn < INF) {
        float s = __expf(mn - a) + __expf(mn - bb) + __expf(mn - cc);
        sm = mn - __logf(s);
      }
      const float4* yj = (const float4*)(Ys + (size_t)(j - 1) * K_);
      float dot = 0.f;
#pragma unroll
      for (int k = 0; k < K_ / 4; ++k) {
        float4 xa = xi[k]; float4 yb = yj[k];
        dot += xa.x * yb.x + xa.y * yb.y + xa.z * yb.z + xa.w * yb.w;
      }
      val = (x2s[t] + y2s[j - 1] - 2.f * dot) + sm;
    }
    buf[pc][i] = val;
    if (t == 0) buf[pc][0] = INF;
    __syncthreads();
    int tmp = p2; p2 = p1; p1 = pc; pc = tmp;
  }
  if (i == N_) out[b] = val;
}

// ---------------------------------------------------------------------------
extern "C" void kernel_launch(void* const* d_in, const int* in_sizes, int n_in,
                              void* d_out, int out_size, void* d_ws, size_t ws_size,
                              hipStream_t stream) {
  const float* X = (const float*)d_in[0];  // [B,N,d] f32
  const float* Y = (const float*)d_in[1];  // [B,M,d] f32
  float* out = (float*)d_out;              // [B] f32

  const size_t dbytes = (size_t)B_ * N_ * M_ * sizeof(float);   // 32 MB
  const size_t nbytes = (size_t)B_ * N_ * sizeof(float);

  if (ws_size >= dbytes + 2 * nbytes) {
    float* D  = (float*)d_ws;
    float* x2 = (float*)((char*)d_ws + dbytes);
    float* y2 = x2 + B_ * N_;

    sdtw_norms<<<(2 * B_ * N_) / 256, 256, 0, stream>>>(X, Y, x2, y2);

    dim3 grid(128, B_);  // 128 blocks * 8 waves = 1024 tiles per batch
    sdtw_pairdist_wmma<<<grid, 256, 0, stream>>>(X, Y, x2, y2, D);

    sdtw_dp<<<B_, N_, 0, stream>>>(D, out);
  } else {
    size_t smem = (size_t)(2 * N_ * K_ + 2 * N_ + 3 * (N_ + 1)) * sizeof(float);
    hipFuncSetAttribute((const void*)sdtw_fused,
                        hipFuncAttributeMaxDynamicSharedMemorySize, (int)smem);
    sdtw_fused<<<B_, N_, smem, stream>>>(X, Y, out);
  }
}